// Net_83580063580622
// MI455X (gfx1250) — compile-verified
//
#include <hip/hip_runtime.h>
#include <hip/hip_bf16.h>

typedef __attribute__((ext_vector_type(2))) float v2f;
typedef __attribute__((ext_vector_type(8))) float v8f;

#define NGRAPH 16384
#define NPG    54
#define NNODE  (NGRAPH * NPG)      // 884736  (multiple of 16)
#define NEDGE  (144 * NGRAPH)      // 2359296
#define FIN    11
#define FHID   16
#define FOUT   3
#define GDIM   10
#define EMB    (NPG * FOUT)        // 162
#define CATK   (EMB + GDIM)        // 172
#define CATKP  176                 // 172 padded to multiple of 16
#define H1DIM  85
#define NPADH  96                  // 85 padded to 6x16
#define YLD    96

// ---------------------------------------------------------------- degree ----
__global__ void k_init_deg(float* deg) {
    int i = blockIdx.x * blockDim.x + threadIdx.x;
    if (i < NNODE) deg[i] = 1.0f;              // self-loop weight 1
}

__global__ void k_deg_scatter(const int* __restrict__ col,
                              const float* __restrict__ ew,
                              float* __restrict__ deg) {
    int e = blockIdx.x * blockDim.x + threadIdx.x;
    if (e < NEDGE) atomicAdd(&deg[col[e]], ew[e]);
}

__global__ void k_deg_to_dis(float* deg_dis) {
    int i = blockIdx.x * blockDim.x + threadIdx.x;
    if (i < NNODE) {
        float d = deg_dis[i];
        deg_dis[i] = d > 0.0f ? rsqrtf(d) : 0.0f;
    }
}

// ------------------------------------------------------ weight pre-packing --
// Zero-pad W[K][N] to [KP][NP] and interleave K-pairs so a lane's B fragment
// (W[ka][n], W[ka+1][n]) with even ka is one contiguous 8-byte load:
//   Wil[((k>>1)*NP + n)*2 + (k&1)] = W[k][n]
template<int K, int N, int KP, int NP>
__global__ void k_pack_w(const float* __restrict__ w, float* __restrict__ wil) {
    int i = blockIdx.x * blockDim.x + threadIdx.x;
    if (i >= KP * NP) return;
    int k = i / NP, n = i % NP;
    float v = (k < K && n < N) ? w[k * N + n] : 0.0f;
    wil[(((k >> 1) * NP) + n) * 2 + (k & 1)] = v;
}

// --------------------------------------------------------- node-GEMM (WMMA) -
// out[m, 0:NSTORE] = A[m, 0:K] @ W[0:K, :]   via V_WMMA_F32_16X16X4_F32
// One wave = 16 node rows. EXEC all-ones through the WMMA loop; all guards
// are value-selects on clamped in-bounds addresses (no divergent branches).
template<int K, int KPAD, int LDA, int NSTORE, bool AGUARD>
__global__ void k_gcn_gemm(const float* __restrict__ A,
                           const float* __restrict__ Wil,   // packed [KPAD/2][16][2]
                           float* __restrict__ out) {
    int wave = (blockIdx.x * blockDim.x + threadIdx.x) >> 5;
    int lane = threadIdx.x & 31;
    int m0   = wave * 16;
    int row  = lane & 15;
    int kh   = (lane >> 4) << 1;        // 0 or 2 (A-layout half select); ka even
    int col  = lane & 15;

    const float* arow = A + (size_t)(m0 + row) * LDA;

    v8f c = {};
    #pragma unroll
    for (int k0 = 0; k0 < KPAD; k0 += 4) {
        int ka = k0 + kh;
        v2f a, b;
        if (AGUARD) {
            int kc0 = (ka     < K) ? ka     : 0;   // clamped, always in-bounds
            int kc1 = (ka + 1 < K) ? ka + 1 : 0;
            float v0 = arow[kc0];
            float v1 = arow[kc1];
            a.x = (ka     < K) ? v0 : 0.0f;
            a.y = (ka + 1 < K) ? v1 : 0.0f;
        } else {
            a = *(const v2f*)(arow + ka);          // 8B aligned (ka even)
        }
        b = *(const v2f*)(Wil + (((ka >> 1) * 16) + col) * 2);
        c = __builtin_amdgcn_wmma_f32_16x16x4_f32(false, a, false, b,
                                                  (short)0, c, false, false);
    }
    int mofs = (lane >> 4) << 3;        // 0 or 8 (C-layout)
    if (col < NSTORE) {
        #pragma unroll
        for (int r = 0; r < 8; ++r)
            out[(size_t)(m0 + mofs + r) * NSTORE + col] = c[r];
    }
}

// ----------------------------------------------------------------- scatter --
template<int F>
__global__ void k_zero(float* __restrict__ acc) {
    size_t i = (size_t)blockIdx.x * blockDim.x + threadIdx.x;
    if (i < (size_t)NNODE * F) acc[i] = 0.0f;
}

__global__ void k_edge_scatter16(const int* __restrict__ rowi,
                                 const int* __restrict__ coli,
                                 const float* __restrict__ ew,
                                 const float* __restrict__ dis,
                                 const float* __restrict__ tmp,
                                 float* __restrict__ acc) {
    int e = blockIdx.x * blockDim.x + threadIdx.x;
    if (e >= NEDGE) return;
    int r = rowi[e], cdst = coli[e];
    float nrm = dis[r] * ew[e] * dis[cdst];
    const float4* src = (const float4*)(tmp + (size_t)r * 16);   // 16B aligned
    float*        dst = acc + (size_t)cdst * 16;
    #pragma unroll
    for (int q = 0; q < 4; ++q) {
        float4 s = src[q];
        atomicAdd(dst + q * 4 + 0, nrm * s.x);
        atomicAdd(dst + q * 4 + 1, nrm * s.y);
        atomicAdd(dst + q * 4 + 2, nrm * s.z);
        atomicAdd(dst + q * 4 + 3, nrm * s.w);
    }
}

__global__ void k_edge_scatter3(const int* __restrict__ rowi,
                                const int* __restrict__ coli,
                                const float* __restrict__ ew,
                                const float* __restrict__ dis,
                                const float* __restrict__ tmp,
                                float* __restrict__ acc) {
    int e = blockIdx.x * blockDim.x + threadIdx.x;
    if (e >= NEDGE) return;
    int r = rowi[e], cdst = coli[e];
    float nrm = dis[r] * ew[e] * dis[cdst];
    const float* src = tmp + (size_t)r * 3;
    float*       dst = acc + (size_t)cdst * 3;
    atomicAdd(dst + 0, nrm * src[0]);
    atomicAdd(dst + 1, nrm * src[1]);
    atomicAdd(dst + 2, nrm * src[2]);
}

// finalize: acc = relu(acc + dis^2 * tmp + b)   (self-loop term folded here)
template<int F>
__global__ void k_finalize(const float* __restrict__ tmp,
                           const float* __restrict__ dis,
                           const float* __restrict__ bias,
                           float* __restrict__ acc) {
    size_t i = (size_t)blockIdx.x * blockDim.x + threadIdx.x;
    if (i >= (size_t)NNODE * F) return;
    int n = (int)(i / F), f = (int)(i % F);
    float d = dis[n];
    float v = acc[i] + d * d * tmp[i] + bias[f];
    acc[i] = fmaxf(v, 0.0f);
}

// --------------------------------------------------------- global MLP -------
__global__ void k_gmlp(const float* __restrict__ gf,
                       const float* __restrict__ Wg1, const float* __restrict__ bg1,
                       const float* __restrict__ Wg2, const float* __restrict__ bg2,
                       const float* __restrict__ Wg3, const float* __restrict__ bg3,
                       float* __restrict__ g) {
    int b = blockIdx.x * blockDim.x + threadIdx.x;
    if (b >= NGRAPH) return;
    float in[GDIM];
    #pragma unroll
    for (int j = 0; j < GDIM; ++j) in[j] = gf[(size_t)b * GDIM + j];
    float h1[16];
    #pragma unroll
    for (int i = 0; i < 16; ++i) {
        float s = bg1[i];
        #pragma unroll
        for (int j = 0; j < GDIM; ++j) s += in[j] * Wg1[j * 16 + i];
        h1[i] = fmaxf(s, 0.0f);
    }
    float h2[16];
    #pragma unroll
    for (int i = 0; i < 16; ++i) {
        float s = bg2[i];
        #pragma unroll
        for (int j = 0; j < 16; ++j) s += h1[j] * Wg2[j * 16 + i];
        h2[i] = fmaxf(s, 0.0f);
    }
    #pragma unroll
    for (int o = 0; o < GDIM; ++o) {
        float s = bg3[o];
        #pragma unroll
        for (int i = 0; i < 16; ++i) s += h2[i] * Wg3[i * GDIM + o];
        g[(size_t)b * GDIM + o] = fmaxf(s, 0.0f);
    }
}

// ---------------------------------------------- concat staging for the head -
// cat[b][0:162] = embeds, [162:172] = g, [172:176] = 0
__global__ void k_cat(const float* __restrict__ h3, const float* __restrict__ g,
                      float* __restrict__ cat) {
    int i = blockIdx.x * blockDim.x + threadIdx.x;
    if (i >= NGRAPH * CATKP) return;
    int b = i / CATKP, k = i % CATKP;
    float v = (k < EMB)  ? h3[(size_t)b * EMB + k]
            : (k < CATK) ? g[(size_t)b * GDIM + (k - EMB)]
            : 0.0f;
    cat[i] = v;
}

// ------------------------------------------------ output head GEMM (WMMA) ---
// y[b, n] = relu( cat[b] @ Wo1 + bo1 ),  n in [0,85)
// One wave: 16 graph rows x six 16-wide N tiles; A fragment loaded once/k-step.
__global__ void k_head_gemm(const float* __restrict__ cat,     // [B][176]
                            const float* __restrict__ Wo1il,   // packed [88][96][2]
                            const float* __restrict__ bo1,
                            float* __restrict__ y) {
    int wave = (blockIdx.x * blockDim.x + threadIdx.x) >> 5;
    int lane = threadIdx.x & 31;
    int b0   = wave * 16;
    int row  = lane & 15;
    int kh   = (lane >> 4) << 1;
    int col  = lane & 15;

    const float* arow = cat + (size_t)(b0 + row) * CATKP;

    v8f acc[6] = {};
    #pragma unroll 4
    for (int k0 = 0; k0 < CATKP; k0 += 4) {
        int ka = k0 + kh;                                  // even
        v2f a = *(const v2f*)(arow + ka);
        const float* wrow = Wo1il + (size_t)(ka >> 1) * NPADH * 2;
        #pragma unroll
        for (int t = 0; t < 6; ++t) {
            v2f b = *(const v2f*)(wrow + (t * 16 + col) * 2);
            acc[t] = __builtin_amdgcn_wmma_f32_16x16x4_f32(false, a, false, b,
                                                           (short)0, acc[t],
                                                           false, false);
        }
    }
    int mofs = (lane >> 4) << 3;
    #pragma unroll
    for (int t = 0; t < 6; ++t) {
        int n = t * 16 + col;
        if (n < H1DIM) {
            #pragma unroll
            for (int r = 0; r < 8; ++r)
                y[(size_t)(b0 + mofs + r) * YLD + n] = fmaxf(acc[t][r] + bo1[n], 0.0f);
        }
    }
}

__global__ void k_head_out(const float* __restrict__ y,
                           const float* __restrict__ Wo2,
                           const float* __restrict__ bo2,
                           float* __restrict__ out) {
    int b = blockIdx.x * blockDim.x + threadIdx.x;
    if (b >= NGRAPH) return;
    float s = bo2[0];
    #pragma unroll 5
    for (int j = 0; j < H1DIM; ++j) s += y[(size_t)b * YLD + j] * Wo2[j];
    out[b] = 1.0f / (1.0f + expf(-s));
}

// ---------------------------------------------------------------- launch ----
extern "C" void kernel_launch(void* const* d_in, const int* in_sizes, int n_in,
                              void* d_out, int out_size, void* d_ws, size_t ws_size,
                              hipStream_t stream) {
    const float* x    = (const float*)d_in[0];
    const int*   ei   = (const int*)  d_in[1];   // [2, E] flat
    const float* ea   = (const float*)d_in[2];
    const float* gf   = (const float*)d_in[3];
    const float* W1   = (const float*)d_in[5];
    const float* b1   = (const float*)d_in[6];
    const float* W2   = (const float*)d_in[7];
    const float* b2   = (const float*)d_in[8];
    const float* W3   = (const float*)d_in[9];
    const float* b3   = (const float*)d_in[10];
    const float* Wg1  = (const float*)d_in[11];
    const float* bg1  = (const float*)d_in[12];
    const float* Wg2  = (const float*)d_in[13];
    const float* bg2  = (const float*)d_in[14];
    const float* Wg3  = (const float*)d_in[15];
    const float* bg3  = (const float*)d_in[16];
    const float* Wo1  = (const float*)d_in[17];
    const float* bo1  = (const float*)d_in[18];
    const float* Wo2  = (const float*)d_in[19];
    const float* bo2  = (const float*)d_in[20];
    float* out = (float*)d_out;

    const int* erow = ei;
    const int* ecol = ei + NEDGE;

    // workspace layout
    float* dis   = (float*)d_ws;                 // NNODE
    float* bufA  = dis   + NNODE;                // NNODE*16
    float* bufB  = bufA  + (size_t)NNODE * 16;   // NNODE*16
    float* bufT  = bufB  + (size_t)NNODE * 16;   // NNODE*16
    float* gbuf  = bufT  + (size_t)NNODE * 16;   // NGRAPH*GDIM
    float* catb  = gbuf  + (size_t)NGRAPH * 16;  // NGRAPH*CATKP
    float* ybuf  = catb  + (size_t)NGRAPH * CATKP; // NGRAPH*YLD
    float* w1p   = ybuf  + (size_t)NGRAPH * YLD; // 12*16
    float* w2p   = w1p   + 12 * 16;              // 16*16
    float* w3p   = w2p   + 16 * 16;              // 16*16
    float* wo1p  = w3p   + 16 * 16;              // 176*96

    const int T = 256;
    const int gN    = (NNODE + T - 1) / T;
    const int gE    = (NEDGE + T - 1) / T;
    const int gN16  = (int)(((size_t)NNODE * 16 + T - 1) / T);
    const int gN3   = (int)(((size_t)NNODE * 3  + T - 1) / T);
    const int gGemm = (NNODE / 16) * 32 / T;      // 6912 blocks, 8 waves each
    const int gHead = (NGRAPH / 16) * 32 / T;     // 128 blocks
    const int gB    = (NGRAPH + T - 1) / T;
    const int gCat  = (NGRAPH * CATKP + T - 1) / T;

    // pack weights (zero-pad + K-pair interleave)
    k_pack_w<FIN,  FHID, 12, 16><<<(12*16 + T-1)/T,   T, 0, stream>>>(W1,  w1p);
    k_pack_w<FHID, FHID, 16, 16><<<(16*16 + T-1)/T,   T, 0, stream>>>(W2,  w2p);
    k_pack_w<FHID, FOUT, 16, 16><<<(16*16 + T-1)/T,   T, 0, stream>>>(W3,  w3p);
    k_pack_w<CATK, H1DIM, CATKP, NPADH><<<(CATKP*NPADH + T-1)/T, T, 0, stream>>>(Wo1, wo1p);

    // degree -> dis
    k_init_deg   <<<gN, T, 0, stream>>>(dis);
    k_deg_scatter<<<gE, T, 0, stream>>>(ecol, ea, dis);
    k_deg_to_dis <<<gN, T, 0, stream>>>(dis);

    // layer 1: x(11) -> bufA(16)
    k_gcn_gemm<FIN, 12, FIN, FHID, true><<<gGemm, T, 0, stream>>>(x, w1p, bufT);
    k_zero<FHID>    <<<gN16, T, 0, stream>>>(bufA);
    k_edge_scatter16<<<gE,   T, 0, stream>>>(erow, ecol, ea, dis, bufT, bufA);
    k_finalize<FHID><<<gN16, T, 0, stream>>>(bufT, dis, b1, bufA);

    // layer 2: bufA(16) -> bufB(16)
    k_gcn_gemm<FHID, 16, FHID, FHID, false><<<gGemm, T, 0, stream>>>(bufA, w2p, bufT);
    k_zero<FHID>    <<<gN16, T, 0, stream>>>(bufB);
    k_edge_scatter16<<<gE,   T, 0, stream>>>(erow, ecol, ea, dis, bufT, bufB);
    k_finalize<FHID><<<gN16, T, 0, stream>>>(bufT, dis, b2, bufB);

    // layer 3: bufB(16) -> bufA(3)   (bufA now holds embeds [B, 162] flat)
    k_gcn_gemm<FHID, 16, FHID, FOUT, false><<<gGemm, T, 0, stream>>>(bufB, w3p, bufT);
    k_zero<FOUT>    <<<gN3, T, 0, stream>>>(bufA);
    k_edge_scatter3 <<<gE,  T, 0, stream>>>(erow, ecol, ea, dis, bufT, bufA);
    k_finalize<FOUT><<<gN3, T, 0, stream>>>(bufT, dis, b3, bufA);

    // global MLP + concat staging
    k_gmlp<<<gB,   T, 0, stream>>>(gf, Wg1, bg1, Wg2, bg2, Wg3, bg3, gbuf);
    k_cat <<<gCat, T, 0, stream>>>(bufA, gbuf, catb);

    // output head
    k_head_gemm<<<gHead, T, 0, stream>>>(catb, wo1p, bo1, ybuf);
    k_head_out <<<gB,    T, 0, stream>>>(ybuf, Wo2, bo2, out);
}